// VarGridEncoder_54778012893617
// MI455X (gfx1250) — compile-verified
//
#include <hip/hip_runtime.h>
#include <stdint.h>

typedef float v4f __attribute__((ext_vector_type(4)));

#define P1 2654435761u
#define P2 805459861u
#define HASH_MASK 524287u      // hashed levels are exactly 2^19 entries
#define NLVL 16
#define EMB_ROWS 4920          // level-0 table == "embeddings" input

__host__ __device__ constexpr unsigned lvl_size(int l) {
    unsigned long long res = 16ull << l;                 // res = 16 * 2^l
    unsigned long long p = (res + 1ull) * (res + 1ull) * (res + 1ull);
    if (p > 524288ull) p = 524288ull;                    // MAXP = 2^19
    p = ((p + 7ull) / 8ull) * 8ull;
    return (unsigned)p;
}
__host__ __device__ constexpr unsigned lvl_off(int l) {
    unsigned o = 0;
    for (int i = 0; i < l; ++i) o += lvl_size(i);
    return o;
}

template <int LVL>
__device__ __forceinline__ float2 level_eval(float x, float y, float z,
                                             const float2* __restrict__ tab,
                                             const float2* s0) {
    constexpr unsigned RES  = 16u << LVL;
    constexpr float    SCALE = (float)(RES - 1u);        // 16*2^l - 1, exact in fp32
    constexpr unsigned S    = RES + 1u;                  // dense row stride
    constexpr bool     DENSE = (LVL <= 2);               // (res+1)^3 <= level size

    // match reference: mul then add, round-to-nearest, no FMA contraction
    float px = __fadd_rn(__fmul_rn(x, SCALE), 0.5f);
    float py = __fadd_rn(__fmul_rn(y, SCALE), 0.5f);
    float pz = __fadd_rn(__fmul_rn(z, SCALE), 0.5f);
    float fx = floorf(px), fy = floorf(py), fz = floorf(pz);
    float tx = px - fx, ty = py - fy, tz = pz - fz;
    unsigned ux = (unsigned)fx, uy = (unsigned)fy, uz = (unsigned)fz;

    float wx0 = 1.0f - tx, wy0 = 1.0f - ty, wz0 = 1.0f - tz;
    float m00 = wy0 * wz0, m10 = ty * wz0, m01 = wy0 * tz, m11 = ty * tz;

    unsigned i0, i1, i2, i3, i4, i5, i6, i7;
    if constexpr (DENSE) {
        unsigned b0 = uy * S,       b1 = b0 + S;
        unsigned c0 = uz * (S * S), c1 = c0 + S * S;
        i0 = ux + b0 + c0; i1 = i0 + 1u;
        i2 = ux + b1 + c0; i3 = i2 + 1u;
        i4 = ux + b0 + c1; i5 = i4 + 1u;
        i6 = ux + b1 + c1; i7 = i6 + 1u;
    } else {
        unsigned a0 = ux,      a1 = ux + 1u;
        unsigned b0 = uy * P1, b1 = b0 + P1;   // uint32 wraparound == numpy
        unsigned c0 = uz * P2, c1 = c0 + P2;
        i0 = (a0 ^ b0 ^ c0) & HASH_MASK; i1 = (a1 ^ b0 ^ c0) & HASH_MASK;
        i2 = (a0 ^ b1 ^ c0) & HASH_MASK; i3 = (a1 ^ b1 ^ c0) & HASH_MASK;
        i4 = (a0 ^ b0 ^ c1) & HASH_MASK; i5 = (a1 ^ b0 ^ c1) & HASH_MASK;
        i6 = (a0 ^ b1 ^ c1) & HASH_MASK; i7 = (a1 ^ b1 ^ c1) & HASH_MASK;
    }

    float r0 = 0.f, r1 = 0.f;
    if constexpr (LVL == 0) {
        // level 0 gathered from LDS (ds_load_b64), staged asynchronously
        float2 e;
        e = s0[i0]; r0 = fmaf(wx0 * m00, e.x, r0); r1 = fmaf(wx0 * m00, e.y, r1);
        e = s0[i1]; r0 = fmaf(tx  * m00, e.x, r0); r1 = fmaf(tx  * m00, e.y, r1);
        e = s0[i2]; r0 = fmaf(wx0 * m10, e.x, r0); r1 = fmaf(wx0 * m10, e.y, r1);
        e = s0[i3]; r0 = fmaf(tx  * m10, e.x, r0); r1 = fmaf(tx  * m10, e.y, r1);
        e = s0[i4]; r0 = fmaf(wx0 * m01, e.x, r0); r1 = fmaf(wx0 * m01, e.y, r1);
        e = s0[i5]; r0 = fmaf(tx  * m01, e.x, r0); r1 = fmaf(tx  * m01, e.y, r1);
        e = s0[i6]; r0 = fmaf(wx0 * m11, e.x, r0); r1 = fmaf(wx0 * m11, e.y, r1);
        e = s0[i7]; r0 = fmaf(tx  * m11, e.x, r0); r1 = fmaf(tx  * m11, e.y, r1);
    } else {
        const float2* __restrict__ base = tab + (lvl_off(LVL) - EMB_ROWS);
        float2 e;
        e = base[i0]; r0 = fmaf(wx0 * m00, e.x, r0); r1 = fmaf(wx0 * m00, e.y, r1);
        e = base[i1]; r0 = fmaf(tx  * m00, e.x, r0); r1 = fmaf(tx  * m00, e.y, r1);
        e = base[i2]; r0 = fmaf(wx0 * m10, e.x, r0); r1 = fmaf(wx0 * m10, e.y, r1);
        e = base[i3]; r0 = fmaf(tx  * m10, e.x, r0); r1 = fmaf(tx  * m10, e.y, r1);
        e = base[i4]; r0 = fmaf(wx0 * m01, e.x, r0); r1 = fmaf(wx0 * m01, e.y, r1);
        e = base[i5]; r0 = fmaf(tx  * m01, e.x, r0); r1 = fmaf(tx  * m01, e.y, r1);
        e = base[i6]; r0 = fmaf(wx0 * m11, e.x, r0); r1 = fmaf(wx0 * m11, e.y, r1);
        e = base[i7]; r0 = fmaf(tx  * m11, e.x, r0); r1 = fmaf(tx  * m11, e.y, r1);
    }
    return make_float2(r0, r1);
}

__global__ __launch_bounds__(256)
void grid_encode(const float* __restrict__ in,
                 const float2* __restrict__ emb,
                 const float2* __restrict__ tab,
                 const int* __restrict__ bound_p,
                 float* __restrict__ out, int n) {
    __shared__ float2 s_lvl0[EMB_ROWS];

    // ---- async stage level-0 table (39360 B) global -> LDS (ASYNCcnt path) ----
    {
        unsigned lds0 = (unsigned)(uintptr_t)(&s_lvl0[0]);  // flat[31:0] == LDS byte addr
        for (unsigned c = threadIdx.x; c < (EMB_ROWS * 8u) / 16u; c += blockDim.x) {
            unsigned off = c * 16u;
            asm volatile("global_load_async_to_lds_b128 %0, %1, %2"
                         :
                         : "v"(lds0 + off), "v"(off), "s"(emb)
                         : "memory");
        }
        asm volatile("s_wait_asynccnt 0x0" ::: "memory");
    }
    __syncthreads();

    const float bf  = (float)(*bound_p);
    const float den = 2.0f * bf;   // bound==1 -> divide by 2, exact

    for (int i = blockIdx.x * blockDim.x + threadIdx.x; i < n;
         i += gridDim.x * blockDim.x) {
        const float* p = in + (size_t)3 * (size_t)i;
        float x = (p[0] + bf) / den;
        float y = (p[1] + bf) / den;
        float z = (p[2] + bf) / den;

        float2 r[NLVL];
        r[0]  = level_eval<0 >(x, y, z, tab, s_lvl0);
        r[1]  = level_eval<1 >(x, y, z, tab, s_lvl0);
        r[2]  = level_eval<2 >(x, y, z, tab, s_lvl0);
        r[3]  = level_eval<3 >(x, y, z, tab, s_lvl0);
        r[4]  = level_eval<4 >(x, y, z, tab, s_lvl0);
        r[5]  = level_eval<5 >(x, y, z, tab, s_lvl0);
        r[6]  = level_eval<6 >(x, y, z, tab, s_lvl0);
        r[7]  = level_eval<7 >(x, y, z, tab, s_lvl0);
        r[8]  = level_eval<8 >(x, y, z, tab, s_lvl0);
        r[9]  = level_eval<9 >(x, y, z, tab, s_lvl0);
        r[10] = level_eval<10>(x, y, z, tab, s_lvl0);
        r[11] = level_eval<11>(x, y, z, tab, s_lvl0);
        r[12] = level_eval<12>(x, y, z, tab, s_lvl0);
        r[13] = level_eval<13>(x, y, z, tab, s_lvl0);
        r[14] = level_eval<14>(x, y, z, tab, s_lvl0);
        r[15] = level_eval<15>(x, y, z, tab, s_lvl0);

        // streaming output: 8x non-temporal b128 stores, don't pollute L2
        v4f* ov = (v4f*)(out + (size_t)i * 32u);
        #pragma unroll
        for (int k = 0; k < 8; ++k) {
            v4f v = {r[2 * k].x, r[2 * k].y, r[2 * k + 1].x, r[2 * k + 1].y};
            __builtin_nontemporal_store(v, ov + k);
        }
    }
}

extern "C" void kernel_launch(void* const* d_in, const int* in_sizes, int n_in,
                              void* d_out, int out_size, void* d_ws, size_t ws_size,
                              hipStream_t stream) {
    const float*  in  = (const float*)d_in[0];
    const float2* emb = (const float2*)d_in[1];   // 4920 x 2 f32 (== level 0)
    const float2* tab = (const float2*)d_in[2];   // remaining levels
    const int*    bnd = (const int*)d_in[3];
    float* out = (float*)d_out;

    int n = in_sizes[0] / 3;
    dim3 grid(2048), block(256);                  // 4 points/thread: amortize LDS staging
    hipLaunchKernelGGL(grid_encode, grid, block, 0, stream,
                       in, emb, tab, bnd, out, n);
    (void)d_ws; (void)ws_size; (void)n_in; (void)out_size;
}